// TripletLoss_10617159156192
// MI455X (gfx1250) — compile-verified
//
#include <hip/hip_runtime.h>

typedef __attribute__((ext_vector_type(2))) float v2f;
typedef __attribute__((ext_vector_type(8))) float v8f;

#define D_FEAT 128
#define T_TRIP 65536
#define WAVES_PER_BLOCK 8
#define TILES (T_TRIP / 16)                    // 4096 tiles of 16 triplets
#define BLOCKS (TILES / WAVES_PER_BLOCK)       // 512 blocks

// One wave32 per tile of 16 triplets.
// Computes diag(U*U^T) and diag(V*V^T) with V_WMMA_F32_16X16X4_F32 where
// U = Xi - Xj, V = Xi - Xk (16x128 fp32), then softplus + reduction.
__global__ __launch_bounds__(256) void triplet_wmma_kernel(
    const float* __restrict__ F,
    const long long* __restrict__ trip,
    float* __restrict__ partial)
{
    __shared__ float wsum[WAVES_PER_BLOCK];

    const int lane = threadIdx.x & 31;
    const int wave = threadIdx.x >> 5;
    const int tile = blockIdx.x * WAVES_PER_BLOCK + wave;

    // Row of the 16-triplet tile this lane serves (two lanes per row).
    const int row = lane & 15;
    const int t   = tile * 16 + row;

    const long long* tp = trip + 3ll * (long long)t;
    const int ia = (int)tp[0];
    const int ja = (int)tp[1];
    const int ka = (int)tp[2];

    const float* fi = F + (size_t)ia * D_FEAT;
    const float* fj = F + (size_t)ja * D_FEAT;
    const float* fk = F + (size_t)ka * D_FEAT;

    // fp32 16x4 A layout: lanes 0-15 hold K={0,1}, lanes 16-31 hold K={2,3}.
    // For B = U^T (4x16) the per-lane register contents are identical, so the
    // same operand registers feed both A and B of the symmetric product.
    const int cb = (lane >> 4) * 2;   // 0 or 2

    v8f cij = {};   // accumulates U * U^T
    v8f cik = {};   // accumulates V * V^T

#pragma unroll 4
    for (int kb = 0; kb < D_FEAT / 4; ++kb) {
        const int c0 = kb * 4 + cb;
        const float2 xi = *(const float2*)(fi + c0);
        const float2 xj = *(const float2*)(fj + c0);
        const float2 xk = *(const float2*)(fk + c0);

        v2f u; u.x = xi.x - xj.x; u.y = xi.y - xj.y;
        v2f v; v.x = xi.x - xk.x; v.y = xi.y - xk.y;

        // 8 args: (neg_a, A, neg_b, B, c_mod, C, reuse_a, reuse_b)
        cij = __builtin_amdgcn_wmma_f32_16x16x4_f32(
            false, u, false, u, (short)0, cij, false, false);
        cik = __builtin_amdgcn_wmma_f32_16x16x4_f32(
            false, v, false, v, (short)0, cik, false, false);
    }

    // Diagonal C[r][r]: r<8 -> (lane r, vgpr r); r>=8 -> (lane 16+r, vgpr r-8).
    float dij = 0.0f, dik = 0.0f;
#pragma unroll
    for (int p = 0; p < 8; ++p) {
        const bool sel = (lane == p) || (lane == 24 + p);
        dij = sel ? cij[p] : dij;
        dik = sel ? cik[p] : dik;
    }
    const bool valid = (lane < 8) || (lane >= 24);

    // Numerically stable softplus(d_ij - d_ik), matching jax.nn.softplus.
    const float x  = dij - dik;
    const float sp = fmaxf(x, 0.0f) + log1pf(expf(-fabsf(x)));
    float contrib  = valid ? sp : 0.0f;

    // Deterministic wave32 reduction.
#pragma unroll
    for (int off = 16; off > 0; off >>= 1)
        contrib += __shfl_xor(contrib, off, 32);

    if (lane == 0) wsum[wave] = contrib;
    __syncthreads();

    if (threadIdx.x == 0) {
        float s = 0.0f;
#pragma unroll
        for (int w = 0; w < WAVES_PER_BLOCK; ++w) s += wsum[w];
        partial[blockIdx.x] = s;
    }
}

// Fixed-order final reduction over BLOCKS (=512) partials -> mean.
__global__ __launch_bounds__(256) void triplet_reduce_kernel(
    const float* __restrict__ partial, float* __restrict__ out)
{
    __shared__ float sm[256];
    const int tid = threadIdx.x;
    sm[tid] = partial[tid] + partial[tid + 256];
    __syncthreads();
#pragma unroll
    for (int off = 128; off > 0; off >>= 1) {
        if (tid < off) sm[tid] += sm[tid + off];
        __syncthreads();
    }
    if (tid == 0) out[0] = sm[0] * (1.0f / (float)T_TRIP);
}

extern "C" void kernel_launch(void* const* d_in, const int* in_sizes, int n_in,
                              void* d_out, int out_size, void* d_ws, size_t ws_size,
                              hipStream_t stream) {
    const float*     F    = (const float*)d_in[0];      // (8192, 128) fp32
    const long long* trip = (const long long*)d_in[1];  // (65536, 3) int64
    float* out     = (float*)d_out;                     // 1 element
    float* partial = (float*)d_ws;                      // BLOCKS floats (2 KB)

    triplet_wmma_kernel<<<BLOCKS, 256, 0, stream>>>(F, trip, partial);
    triplet_reduce_kernel<<<1, 256, 0, stream>>>(partial, out);
}